// FeatureEmbeddings_62912680952639
// MI455X (gfx1250) — compile-verified
//
#include <hip/hip_runtime.h>
#include <hip/hip_bf16.h>

typedef float v2f __attribute__((ext_vector_type(2)));
typedef float v8f __attribute__((ext_vector_type(8)));

#define HH   64    // hidden dim (K)
#define DD   128   // d_model  (N)
#define BC   128   // batch rows per workgroup (8 waves x 16)
#define KPAD 68    // padded K-stride in LDS: 68 mod 64 = 4 dwords -> conflict-free frag reads

__launch_bounds__(256, 2)
__global__ void feat_mlp_wmma(const float* __restrict__ x,
                              const float* __restrict__ W1,
                              const float* __restrict__ b1,
                              const float* __restrict__ W2,
                              const float* __restrict__ b2,
                              float* __restrict__ out,
                              int F) {
    __shared__ float h_s[BC][KPAD];   // A tile: [row][k], relu(x*W1+b1)
    __shared__ float w2t[DD][KPAD];   // B tile transposed: [col][k] = W2[f][k][col]
    __shared__ float x_s[BC];
    __shared__ float w1_s[HH];
    __shared__ float b1_s[HH];
    __shared__ float b2_s[DD];

    const int f    = blockIdx.x;          // feature index
    const int rb   = blockIdx.y * BC;     // batch row base
    const int tid  = threadIdx.x;
    const int lane = tid & 31;
    const int wave = tid >> 5;

    // ---- phase 0: stage inputs into LDS ----
    if (tid < BC) x_s[tid]  = x[(size_t)(rb + tid) * F + f];
    if (tid < HH) { w1_s[tid] = W1[(size_t)f * HH + tid];
                    b1_s[tid] = b1[(size_t)f * HH + tid]; }
    if (tid < DD) b2_s[tid] = b2[(size_t)f * DD + tid];

    {   // W2[f] (64x128) -> LDS transposed as w2t[d][h].
        // Mapping: h = idx&63 varies across lanes in a wave -> LDS store banks
        // (h + 4*(d)) mod 64 are all distinct -> conflict-free transpose stores.
        const float4* w2g = (const float4*)(W2 + (size_t)f * HH * DD);
        #pragma unroll
        for (int i = 0; i < (HH * DD / 4) / 256; ++i) {
            const int idx  = tid + i * 256;
            const int h    = idx & (HH - 1);
            const int dblk = idx >> 6;            // 0..31, picks float4 within row h
            const float4 v = w2g[h * (DD / 4) + dblk];
            const int d    = dblk * 4;
            w2t[d + 0][h] = v.x;
            w2t[d + 1][h] = v.y;
            w2t[d + 2][h] = v.z;
            w2t[d + 3][h] = v.w;
        }
    }
    __syncthreads();

    // ---- phase 1: hidden tile h = relu(x*W1 + b1), 128x64 ----
    #pragma unroll
    for (int i = 0; i < (BC * HH) / 256; ++i) {
        const int idx = tid + i * 256;
        const int r = idx >> 6;
        const int h = idx & (HH - 1);
        const float hv = fmaf(x_s[r], w1_s[h], b1_s[h]);
        h_s[r][h] = fmaxf(hv, 0.0f);
    }
    __syncthreads();

    // ---- phase 2: per-wave 16x128 strip via v_wmma_f32_16x16x4_f32 ----
    const int m0  = wave * 16;        // row tile of this wave
    const int lo  = lane & 15;        // M for A-frag, N for B-frag
    const int kof = (lane >> 4) * 2;  // K sub-offset within 4-wide K slab

    const float* aRow = &h_s[m0 + lo][0];

    v8f acc[8];
    {
        v8f zero = {};
        #pragma unroll
        for (int nt = 0; nt < 8; ++nt) acc[nt] = zero;
    }

    #pragma unroll 4
    for (int k0 = 0; k0 < HH; k0 += 4) {
        // A fragment (16x4): contiguous 8B in LDS -> single ds_load_b64
        const v2f a = *(const v2f*)(aRow + k0 + kof);
        #pragma unroll
        for (int nt = 0; nt < 8; ++nt) {
            // B fragment (4x16): w2t[col][k..k+1] contiguous 8B -> ds_load_b64
            const v2f bf = *(const v2f*)(&w2t[nt * 16 + lo][k0 + kof]);
            acc[nt] = __builtin_amdgcn_wmma_f32_16x16x4_f32(
                /*neg_a=*/false, a, /*neg_b=*/false, bf,
                /*c_mod=*/(short)0, acc[nt],
                /*reuse_a=*/false, /*reuse_b=*/false);
        }
    }

    // ---- epilogue: +b2, relu, non-temporal store ----
    // C/D layout: VGPR v, lanes 0-15 -> row v, lanes 16-31 -> row v+8; col = lane&15
    const int mhi = (lane >> 4) * 8;
    #pragma unroll
    for (int nt = 0; nt < 8; ++nt) {
        const int col  = nt * 16 + lo;
        const float bb = b2_s[col];
        #pragma unroll
        for (int v = 0; v < 8; ++v) {
            const size_t row = (size_t)(rb + m0 + mhi + v);
            float val = acc[nt][v] + bb;
            val = fmaxf(val, 0.0f);
            __builtin_nontemporal_store(val, &out[(row * F + f) * DD + col]);
        }
    }
}

extern "C" void kernel_launch(void* const* d_in, const int* in_sizes, int n_in,
                              void* d_out, int out_size, void* d_ws, size_t ws_size,
                              hipStream_t stream) {
    const float* x  = (const float*)d_in[0];
    const float* W1 = (const float*)d_in[1];
    const float* b1 = (const float*)d_in[2];
    const float* W2 = (const float*)d_in[3];
    const float* b2 = (const float*)d_in[4];
    float* out = (float*)d_out;

    const int F = in_sizes[4] / DD;   // b2 is (F, D) -> 700
    const int B = in_sizes[0] / F;    // x  is (B, F) -> 512

    dim3 grid(F, B / BC);             // 700 x 4 workgroups
    feat_mlp_wmma<<<grid, 256, 0, stream>>>(x, W1, b1, W2, b2, out, F);
}